// gcn_net_59725815218261
// MI455X (gfx1250) — compile-verified
//
#include <hip/hip_runtime.h>

typedef __attribute__((ext_vector_type(2))) float v2f;
typedef __attribute__((ext_vector_type(8))) float v8f;

#define WAVES_PER_BLOCK 8

// ---------------- degree / dinv ----------------
__global__ void k_deg_init(float* __restrict__ deg, int N) {
  int i = blockIdx.x * blockDim.x + threadIdx.x;
  if (i < N) deg[i] = 1.0f;  // self-loop contributes 1
}

__global__ void k_deg_acc(const int* __restrict__ dst, float* __restrict__ deg, int E) {
  int e = blockIdx.x * blockDim.x + threadIdx.x;
  if (e < E) atomicAdd(&deg[dst[e]], 1.0f);
}

__global__ void k_rsqrt(float* __restrict__ d, int N) {
  int i = blockIdx.x * blockDim.x + threadIdx.x;
  if (i < N) d[i] = rsqrtf(d[i]);  // deg >= 1 always
}

// ---------------- collapse fc_gcn + fc1 into 2x2 ----------------
// Wc = W_gcn(2x32) @ W1(32x2);  bc = b_gcn @ W1 + b1
__global__ void k_make_wc(const float* __restrict__ Wg, const float* __restrict__ bg,
                          const float* __restrict__ W1, const float* __restrict__ b1,
                          float* __restrict__ wcbc) {
  if (threadIdx.x == 0 && blockIdx.x == 0) {
    for (int i = 0; i < 2; ++i)
      for (int j = 0; j < 2; ++j) {
        float s = 0.f;
        for (int k = 0; k < 32; ++k) s += Wg[i * 32 + k] * W1[k * 2 + j];
        wcbc[i * 2 + j] = s;
      }
    for (int j = 0; j < 2; ++j) {
      float s = b1[j];
      for (int k = 0; k < 32; ++k) s += bg[k] * W1[k * 2 + j];
      wcbc[4 + j] = s;
    }
  }
}

// ---------------- transpose to node-major + pre-scale by dinv ----------------
// x[b][c][t][n] * dinv[n] -> xs[n][b][t][c]; acc initialized with self-loop term
__global__ void k_xform(const float* __restrict__ x, const float* __restrict__ dinv,
                        float* __restrict__ xs, float* __restrict__ acc,
                        int N, int total) {
  int idx = blockIdx.x * blockDim.x + threadIdx.x;
  if (idx >= total) return;
  int n = idx % N;
  int r = idx / N;          // (b*2 + c)*12 + t
  int t = r % 12;
  int c = (r / 12) & 1;
  int b = r / 24;
  float v = x[idx] * dinv[n];
  size_t o = (size_t)(n * 8 + b) * 24 + t * 2 + c;
  xs[o]  = v;
  acc[o] = v;
}

// ---------------- edge scatter: acc[dst] += xs[src] (192 floats/edge) --------
// one 192-thread block per edge: coalesced 768B gather + 768B atomic add
__global__ void k_edge_scatter(const int* __restrict__ src, const int* __restrict__ dst,
                               const float* __restrict__ xs, float* __restrict__ acc) {
  int e = blockIdx.x;
  int j = threadIdx.x;      // 0..191 == (b*12+t)*2+c
  int s = src[e];
  int d = dst[e];
  atomicAdd(&acc[(size_t)d * 192 + j], xs[(size_t)s * 192 + j]);
}

// ---------------- fused fc1(2x2)+relu -> fc2(12x64 via WMMA)+relu -> fc3 -----
// Each wave: 16 rows (16 consecutive nodes, fixed b,c). A = 16x12 activations,
// B = 12x64 of W2, 3 K-steps x 4 N-tiles = 12 v_wmma_f32_16x16x4_f32 per wave.
// W2|b2|W3 (896 floats) staged in LDS once per block via async global->LDS DMA.
__global__ __launch_bounds__(256) void k_stageB_wmma(
    const float* __restrict__ acc, const float* __restrict__ dinv,
    const float* __restrict__ wcbc,
    const float* __restrict__ W2, const float* __restrict__ b2,
    const float* __restrict__ W3, const float* __restrict__ b3,
    float* __restrict__ out, int N, int nrows) {
  __shared__ float smem[896];  // [0,768): W2, [768,832): b2, [832,896): W3

  const int tid = threadIdx.x;
  // 224 lanes x 16B = 3584B: one async b128 transfer each (ASYNCcnt path)
  if (tid < 224) {
    const float* gsrc = (tid < 192) ? (W2 + tid * 4)
                      : (tid < 208) ? (b2 + (tid - 192) * 4)
                                    : (W3 + (tid - 208) * 4);
    unsigned ldsOff = (unsigned)(uintptr_t)(&smem[tid * 4]);  // addr[31:0] = LDS offset
    asm volatile("global_load_async_to_lds_b128 %0, %1, off"
                 :: "v"(ldsOff), "v"(gsrc) : "memory");
  }
  asm volatile("s_wait_asynccnt 0x0" ::: "memory");
  __syncthreads();

  const int lane = tid & 31;
  const int wave = blockIdx.x * WAVES_PER_BLOCK + (tid >> 5);
  const int row0 = wave * 16;                 // global row = (b*2+c)*N + n
  if (row0 >= nrows) return;                  // wave-uniform: EXEC stays all-1s

  const int bc = row0 / N;                    // same for all 16 rows of the wave
  const int n0 = row0 - bc * N;
  const int c  = bc & 1;
  const int b  = bc >> 1;

  const int half = lane >> 4;                 // K-half per ISA A-layout
  const int ln   = lane & 15;                 // row (for A) / column (for B,C,D)
  const int n    = n0 + ln;

  const float wc0 = wcbc[c];                  // Wc[0][c]
  const float wc1 = wcbc[2 + c];              // Wc[1][c]
  const float bcc = wcbc[4 + c];
  const float dn  = dinv[n];

  const float* arow = acc + (size_t)(n * 8 + b) * 24;  // [t][c] pairs

  // A fragments: lane(half,ln) holds A[M=ln][K=2*half + {0,1}] for each K-step.
  v2f afrag[3];
#pragma unroll
  for (int kk = 0; kk < 3; ++kk) {
    const int t0 = 4 * kk + 2 * half;
    const float2 p0 = *(const float2*)(arow + t0 * 2);
    const float2 p1 = *(const float2*)(arow + t0 * 2 + 2);
    afrag[kk].x = fmaxf(dn * (p0.x * wc0 + p0.y * wc1) + bcc, 0.f);
    afrag[kk].y = fmaxf(dn * (p1.x * wc0 + p1.y * wc1) + bcc, 0.f);
  }

  float pv[8];
#pragma unroll
  for (int v = 0; v < 8; ++v) pv[v] = 0.f;

#pragma unroll
  for (int ct = 0; ct < 4; ++ct) {            // 4 column tiles of W2 (64 cols)
    const int col = 16 * ct + ln;
    const float bias = smem[768 + col];
    v8f dacc;
#pragma unroll
    for (int v = 0; v < 8; ++v) dacc[v] = bias;   // C[M][col] = b2[col] for all M
#pragma unroll
    for (int kk = 0; kk < 3; ++kk) {
      const int trow = 4 * kk + 2 * half;
      v2f bfrag;                               // B[K=2*half+{0,1}][N=ln]
      bfrag.x = smem[trow * 64 + col];
      bfrag.y = smem[trow * 64 + 64 + col];
      dacc = __builtin_amdgcn_wmma_f32_16x16x4_f32(
          /*neg_a=*/false, afrag[kk], /*neg_b=*/false, bfrag,
          /*c_mod=*/(short)0, dacc, /*reuse_a=*/false, /*reuse_b=*/false);
    }
    // ReLU then fc3 partial: pv[v] accumulates row (v + 8*half) over this column
    const float w3v = smem[832 + col];
#pragma unroll
    for (int v = 0; v < 8; ++v) pv[v] += fmaxf(dacc[v], 0.f) * w3v;
  }

  // reduce over the 16 lanes of each half-wave (columns of the D tile)
#pragma unroll
  for (int m = 1; m < 16; m <<= 1) {
#pragma unroll
    for (int v = 0; v < 8; ++v) pv[v] += __shfl_xor(pv[v], m, 32);
  }

  // row (v + 8*half) written by lane ln==v of that half
  if (ln < 8) {
    float res = pv[0];
#pragma unroll
    for (int v = 1; v < 8; ++v)
      if (ln == v) res = pv[v];
    out[row0 + half * 8 + ln] = res + b3[0];
  }
}

// ---------------- host ----------------
extern "C" void kernel_launch(void* const* d_in, const int* in_sizes, int n_in,
                              void* d_out, int out_size, void* d_ws, size_t ws_size,
                              hipStream_t stream) {
  const float* x  = (const float*)d_in[0];
  const int*   ei = (const int*)d_in[1];
  const float* Wg = (const float*)d_in[2];
  const float* bg = (const float*)d_in[3];
  const float* W1 = (const float*)d_in[4];
  const float* b1 = (const float*)d_in[5];
  const float* W2 = (const float*)d_in[6];
  const float* b2 = (const float*)d_in[7];
  const float* W3 = (const float*)d_in[8];
  const float* b3 = (const float*)d_in[9];
  float* out = (float*)d_out;

  const int Bb = 8, C = 2, T = 12;
  const int total = in_sizes[0];
  const int N = total / (Bb * C * T);    // 20000
  const int E = in_sizes[1] / 2;         // 640000
  const int BTC = Bb * T * C;            // 192

  // workspace layout (floats): dinv[N] | wcbc[16] | xs[N*192] | acc[N*192]
  float* ws   = (float*)d_ws;
  float* dinv = ws;
  size_t Na   = (size_t)((N + 15) / 16) * 16;
  float* wcbc = ws + Na;
  float* xs   = wcbc + 16;
  float* accb = xs + (size_t)N * BTC;
  (void)n_in; (void)out_size; (void)ws_size;

  k_deg_init<<<(N + 255) / 256, 256, 0, stream>>>(dinv, N);
  k_deg_acc<<<(E + 255) / 256, 256, 0, stream>>>(ei + E, dinv, E);
  k_rsqrt<<<(N + 255) / 256, 256, 0, stream>>>(dinv, N);
  k_make_wc<<<1, 32, 0, stream>>>(Wg, bg, W1, b1, wcbc);
  k_xform<<<(total + 255) / 256, 256, 0, stream>>>(x, dinv, xs, accb, N, total);
  k_edge_scatter<<<E, BTC, 0, stream>>>(ei, ei + E, xs, accb);

  const int nrows = Bb * C * N;                       // 320000
  const int rows_per_block = WAVES_PER_BLOCK * 16;    // 128
  k_stageB_wmma<<<(nrows + rows_per_block - 1) / rows_per_block, 256, 0, stream>>>(
      accb, dinv, wcbc, W2, b2, W3, b3, out, N, nrows);
}